// TAGC_4913442587089
// MI455X (gfx1250) — compile-verified
//
#include <hip/hip_runtime.h>
#include <hip/hip_bf16.h>
#include <math.h>

#define N_NODES 100000
#define N_EDGES 1600000
#define FEATD   16
#define HIDD    32
#define IDED    16
#define EMBD    8
#define IN_DIM  64   // IDE + HID + 2*EMB
#define NCD     2
#define LN_EPS  1e-5f

typedef float v2f __attribute__((ext_vector_type(2)));
typedef float v8f __attribute__((ext_vector_type(8)));

__device__ __forceinline__ float elu1(float v) { return v > 0.f ? v : expm1f(v); }

// ---------------------------------------------------------------------------
// Stage 1: per-node input MLPs + concat + LayerNorm -> feat (N x 64)
// ---------------------------------------------------------------------------
__global__ void node_feat_kernel(const float* __restrict__ x,
                                 const int*   __restrict__ cat,
                                 const float* __restrict__ id_table,
                                 const float* __restrict__ Wid,  const float* __restrict__ bid,
                                 const float* __restrict__ emb1, const float* __restrict__ emb2,
                                 const float* __restrict__ Wemb, const float* __restrict__ bemb,
                                 const float* __restrict__ W0,   const float* __restrict__ b0,
                                 const float* __restrict__ ln_g, const float* __restrict__ ln_b,
                                 float* __restrict__ feat)
{
    int i = blockIdx.x * blockDim.x + threadIdx.x;
    if (i >= N_NODES) return;

    float f[IN_DIM];

    // --- ide = elu(id_table[cat0] @ Wid + bid) -> f[0..15]
    {
        const float* idr = id_table + (long)cat[i * 3 + 0] * IDED;
        float t[IDED];
        #pragma unroll
        for (int k = 0; k < IDED; ++k) t[k] = idr[k];
        #pragma unroll
        for (int j = 0; j < IDED; ++j) {
            float s = bid[j];
            #pragma unroll
            for (int k = 0; k < IDED; ++k) s += t[k] * Wid[k * IDED + j];
            f[j] = elu1(s);
        }
    }

    // --- h = elu(x @ W0 + b0) -> f[16..47]
    {
        float t[FEATD];
        #pragma unroll
        for (int k = 0; k < FEATD; ++k) t[k] = x[(long)i * FEATD + k];
        #pragma unroll
        for (int j = 0; j < HIDD; ++j) {
            float s = b0[j];
            #pragma unroll
            for (int k = 0; k < FEATD; ++k) s += t[k] * W0[k * HIDD + j];
            f[IDED + j] = elu1(s);
        }
    }

    // --- er = elu([emb1[cat1], emb2[cat2]] @ Wemb + bemb) -> f[48..63]
    {
        float t[2 * EMBD];
        const float* e1 = emb1 + (long)cat[i * 3 + 1] * EMBD;
        const float* e2 = emb2 + (long)cat[i * 3 + 2] * EMBD;
        #pragma unroll
        for (int k = 0; k < EMBD; ++k) { t[k] = e1[k]; t[EMBD + k] = e2[k]; }
        #pragma unroll
        for (int j = 0; j < 2 * EMBD; ++j) {
            float s = bemb[j];
            #pragma unroll
            for (int k = 0; k < 2 * EMBD; ++k) s += t[k] * Wemb[k * (2 * EMBD) + j];
            f[IDED + HIDD + j] = elu1(s);
        }
    }

    // --- LayerNorm over 64 dims
    float m = 0.f;
    #pragma unroll
    for (int d = 0; d < IN_DIM; ++d) m += f[d];
    m *= (1.f / IN_DIM);
    float v = 0.f;
    #pragma unroll
    for (int d = 0; d < IN_DIM; ++d) { float c = f[d] - m; v += c * c; }
    v *= (1.f / IN_DIM);
    float inv = rsqrtf(v + LN_EPS);
    #pragma unroll
    for (int d = 0; d < IN_DIM; ++d)
        feat[(long)i * IN_DIM + d] = (f[d] - m) * inv * ln_g[d] + ln_b[d];
}

// ---------------------------------------------------------------------------
// Stage 2: degree / normalization
// ---------------------------------------------------------------------------
__global__ void deg_kernel(const int* __restrict__ ei, const float* __restrict__ ew,
                           float* __restrict__ deg)
{
    int e = blockIdx.x * blockDim.x + threadIdx.x;
    if (e >= N_EDGES) return;
    atomicAdd(&deg[ei[N_EDGES + e]], ew[e]);   // col = ei[1][e]
}

__global__ void dis_kernel(float* __restrict__ deg)
{
    int i = blockIdx.x * blockDim.x + threadIdx.x;
    if (i >= N_NODES) return;
    float d = deg[i];
    deg[i] = (d > 0.f) ? rsqrtf(d) : 0.f;
}

__global__ void norm_kernel(const int* __restrict__ ei, const float* __restrict__ ew,
                            const float* __restrict__ dis, float* __restrict__ nrm)
{
    int e = blockIdx.x * blockDim.x + threadIdx.x;
    if (e >= N_EDGES) return;
    nrm[e] = dis[ei[e]] * ew[e] * dis[ei[N_EDGES + e]];
}

// ---------------------------------------------------------------------------
// Stage 3: weighted scatter-sum over edges. 16 threads per edge, float4 lanes,
// f32 global atomics (bandwidth-bound stage: ~820MB/hop at 23.3TB/s).
// ---------------------------------------------------------------------------
__global__ void scatter_kernel(const float* __restrict__ src, float* __restrict__ dst,
                               const int* __restrict__ ei, const float* __restrict__ nrm)
{
    long tid = (long)blockIdx.x * blockDim.x + threadIdx.x;
    long e = tid >> 4;
    if (e >= N_EDGES) return;
    int d = ((int)tid & 15) * 4;
    int r = ei[e];
    int c = ei[N_EDGES + e];
    float w = nrm[e];
    const float4 v = *(const float4*)(src + (long)r * IN_DIM + d);
    float* dp = dst + (long)c * IN_DIM + d;
    atomicAdd(dp + 0, v.x * w);
    atomicAdd(dp + 1, v.y * w);
    atomicAdd(dp + 2, v.z * w);
    atomicAdd(dp + 3, v.w * w);
}

// ---------------------------------------------------------------------------
// Stage 4: Out += X(Nx64) @ W(64x32) using V_WMMA_F32_16X16X4_F32.
// One wave = 16-node tile x 32 cols (two 16x16 accumulators), K in steps of 4.
// ---------------------------------------------------------------------------
__global__ void tag_gemm_wmma(const float* __restrict__ X, const float* __restrict__ W,
                              float* __restrict__ Out, int accumulate)
{
    int lane = threadIdx.x & 31;
    int wave = threadIdx.x >> 5;
    int tile = blockIdx.x * (blockDim.x >> 5) + wave;
    if (tile * 16 >= N_NODES) return;   // wave-uniform guard: EXEC stays all-1s

    int r  = lane & 15;   // row-within-tile for A, col-within-ntile for B/C/D
    int kh = lane >> 4;   // K-half selector for A/B fragments
    long row = (long)tile * 16 + r;

    v8f acc0 = {0.f, 0.f, 0.f, 0.f, 0.f, 0.f, 0.f, 0.f};
    v8f acc1 = {0.f, 0.f, 0.f, 0.f, 0.f, 0.f, 0.f, 0.f};
    if (accumulate) {
        #pragma unroll
        for (int v = 0; v < 8; ++v) {
            long m = (long)tile * 16 + v + 8 * kh;      // C/D layout: M = v + 8*(lane>=16)
            acc0[v] = Out[m * HIDD + r];
            acc1[v] = Out[m * HIDD + r + 16];
        }
    }

    #pragma unroll
    for (int kk = 0; kk < IN_DIM; kk += 4) {
        // A 16x4 f32: a[v] = A[r][kk + v + 2*kh]
        v2f a;
        a.x = X[row * IN_DIM + kk + 0 + 2 * kh];
        a.y = X[row * IN_DIM + kk + 1 + 2 * kh];
        // B 4x16 f32 per N-tile: b[v] = W[kk + v + 2*kh][col]
        v2f b0, b1;
        b0.x = W[(kk + 0 + 2 * kh) * HIDD + r];
        b0.y = W[(kk + 1 + 2 * kh) * HIDD + r];
        b1.x = W[(kk + 0 + 2 * kh) * HIDD + r + 16];
        b1.y = W[(kk + 1 + 2 * kh) * HIDD + r + 16];
        acc0 = __builtin_amdgcn_wmma_f32_16x16x4_f32(false, a, false, b0,
                                                     (short)0, acc0, false, false);
        acc1 = __builtin_amdgcn_wmma_f32_16x16x4_f32(false, a, false, b1,
                                                     (short)0, acc1, false, false);
    }

    #pragma unroll
    for (int v = 0; v < 8; ++v) {
        long m = (long)tile * 16 + v + 8 * kh;
        Out[m * HIDD + r]      = acc0[v];
        Out[m * HIDD + r + 16] = acc1[v];
    }
}

// ---------------------------------------------------------------------------
// Stage 5: bias + relu + LN + classifier + log_softmax (NC=2)
// ---------------------------------------------------------------------------
__global__ void final_kernel(const float* __restrict__ O, const float* __restrict__ tb,
                             const float* __restrict__ g, const float* __restrict__ b,
                             const float* __restrict__ W1, const float* __restrict__ b1,
                             float* __restrict__ out)
{
    int i = blockIdx.x * blockDim.x + threadIdx.x;
    if (i >= N_NODES) return;
    float h[HIDD];
    float m = 0.f;
    #pragma unroll
    for (int j = 0; j < HIDD; ++j) {
        float v = O[(long)i * HIDD + j] + tb[j];
        h[j] = fmaxf(v, 0.f);
        m += h[j];
    }
    m *= (1.f / HIDD);
    float var = 0.f;
    #pragma unroll
    for (int j = 0; j < HIDD; ++j) { float c = h[j] - m; var += c * c; }
    var *= (1.f / HIDD);
    float inv = rsqrtf(var + LN_EPS);
    float l0 = b1[0], l1 = b1[1];
    #pragma unroll
    for (int j = 0; j < HIDD; ++j) {
        float hn = (h[j] - m) * inv * g[j] + b[j];
        l0 += hn * W1[j * NCD + 0];
        l1 += hn * W1[j * NCD + 1];
    }
    float mx = fmaxf(l0, l1);
    float lse = mx + logf(expf(l0 - mx) + expf(l1 - mx));
    out[(long)i * NCD + 0] = l0 - lse;
    out[(long)i * NCD + 1] = l1 - lse;
}

// ---------------------------------------------------------------------------
extern "C" void kernel_launch(void* const* d_in, const int* in_sizes, int n_in,
                              void* d_out, int out_size, void* d_ws, size_t ws_size,
                              hipStream_t stream)
{
    const float* x        = (const float*)d_in[0];
    const int*   ei       = (const int*)  d_in[1];
    const float* ew       = (const float*)d_in[2];
    const int*   cat      = (const int*)  d_in[3];
    const float* id_table = (const float*)d_in[4];
    const float* Wid      = (const float*)d_in[5];
    const float* bid      = (const float*)d_in[6];
    const float* emb1     = (const float*)d_in[7];
    const float* emb2     = (const float*)d_in[8];
    const float* Wemb     = (const float*)d_in[9];
    const float* bemb     = (const float*)d_in[10];
    const float* W0       = (const float*)d_in[11];
    const float* b0       = (const float*)d_in[12];
    const float* ln0_g    = (const float*)d_in[13];
    const float* ln0_b    = (const float*)d_in[14];
    const float* tagW     = (const float*)d_in[15];
    const float* tag_b    = (const float*)d_in[16];
    const float* ln1_g    = (const float*)d_in[17];
    const float* ln1_b    = (const float*)d_in[18];
    const float* W1       = (const float*)d_in[19];
    const float* b1       = (const float*)d_in[20];

    // workspace layout (floats)
    float* feat   = (float*)d_ws;                               // N*64
    float* buf    = feat   + (size_t)N_NODES * IN_DIM;          // N*64
    float* outacc = buf    + (size_t)N_NODES * IN_DIM;          // N*32
    float* deg    = outacc + (size_t)N_NODES * HIDD;            // N
    float* nrm    = deg    + (size_t)N_NODES;                   // E

    // Stage 1: node features
    node_feat_kernel<<<(N_NODES + 127) / 128, 128, 0, stream>>>(
        x, cat, id_table, Wid, bid, emb1, emb2, Wemb, bemb, W0, b0, ln0_g, ln0_b, feat);

    // Stage 2: degree-based edge normalization
    hipMemsetAsync(deg, 0, (size_t)N_NODES * sizeof(float), stream);
    deg_kernel <<<(N_EDGES + 255) / 256, 256, 0, stream>>>(ei, ew, deg);
    dis_kernel <<<(N_NODES + 255) / 256, 256, 0, stream>>>(deg);
    norm_kernel<<<(N_EDGES + 255) / 256, 256, 0, stream>>>(ei, ew, deg, nrm);

    // out = feat @ tag_W[0]
    const int gemm_blocks = (N_NODES / 16 + 7) / 8;   // 8 waves/block, 1 tile/wave
    tag_gemm_wmma<<<gemm_blocks, 256, 0, stream>>>(feat, tagW, outacc, 0);

    // K hops: xk = scatter(xk, norm); out += xk @ tag_W[k]
    float* cur = feat;
    float* nxt = buf;
    for (int k = 1; k <= 3; ++k) {
        hipMemsetAsync(nxt, 0, (size_t)N_NODES * IN_DIM * sizeof(float), stream);
        long total = (long)N_EDGES * 16;
        scatter_kernel<<<(int)((total + 255) / 256), 256, 0, stream>>>(cur, nxt, ei, nrm);
        tag_gemm_wmma<<<gemm_blocks, 256, 0, stream>>>(nxt, tagW + (size_t)k * IN_DIM * HIDD,
                                                       outacc, 1);
        float* t = cur; cur = nxt; nxt = t;
    }

    // Stage 5: head
    final_kernel<<<(N_NODES + 255) / 256, 256, 0, stream>>>(
        outacc, tag_b, ln1_g, ln1_b, W1, b1, (float*)d_out);
}